// FocalLoss_74217034875114
// MI455X (gfx1250) — compile-verified
//
#include <hip/hip_runtime.h>
#include <hip/hip_bf16.h>
#include <cmath>

// ---------------------------------------------------------------------------
// RetinaNet-style focal loss for MI455X (gfx1250, wave32).
// Phase 0: per-annotation precompute (box2d, VP targets, norms) -> ws "aux"
// Phase 1: per-anchor IoU argmax vs 128 boxes; annotation table AND anchor
//          chunk staged in LDS via two TDM descriptors (tensor_load_to_lds),
//          cls rows prefetched (global_prefetch_b8); focal/reg/vp accumulate,
//          deterministic block reduce
// Phase 2: fixed-order finalize -> 3 floats
// ---------------------------------------------------------------------------

namespace fl {

constexpr int B_ = 8;
constexpr int A_ = 100000;
constexpr int C_ = 8;
constexpr int M_ = 128;

constexpr int TPB   = 256;          // 8 wave32 per block
constexpr int APT   = 4;            // anchors per thread
constexpr int CHUNK = TPB * APT;    // 1024 anchors per block
constexpr int NCHUNK = (A_ + CHUNK - 1) / CHUNK;   // 98

constexpr int REC = 32;             // floats per annotation record (128 B)
// record layout: [0..15] coords, [16..18] tx1,ty1,n1, [19..21] tx2,ty2,n2,
//                [22..24] tx3,ty3,n3, [25] label, [26] valid, [27] pad,
//                [28..31] box2d (16B aligned for ds_load_b128)

typedef unsigned int uint32x4 __attribute__((ext_vector_type(4)));
typedef int          int32x4  __attribute__((ext_vector_type(4)));
typedef int          int32x8  __attribute__((ext_vector_type(8)));

#if defined(__has_builtin)
#if __has_builtin(__builtin_amdgcn_tensor_load_to_lds)
#define FL_HAVE_TDM 1
#endif
#endif
#ifndef FL_HAVE_TDM
#define FL_HAVE_TDM 0
#endif

// ---------------------------------------------------------------------------
// Phase 0: one thread per (b, m) annotation.
// ---------------------------------------------------------------------------
__global__ void phase0_kernel(const float* __restrict__ ann,
                              float* __restrict__ aux) {
  int i = blockIdx.x * blockDim.x + threadIdx.x;
  if (i >= B_ * M_) return;
  const float* a = ann + (size_t)i * 18;
  float t[16];
#pragma unroll
  for (int j = 0; j < 16; ++j) t[j] = a[j];
  float label = a[17];
  bool valid = (label >= 0.0f);

  float minx = t[0], maxx = t[0], miny = t[1], maxy = t[1];
#pragma unroll
  for (int j = 1; j < 8; ++j) {
    minx = fminf(minx, t[2 * j]);
    maxx = fmaxf(maxx, t[2 * j]);
    miny = fminf(miny, t[2 * j + 1]);
    maxy = fmaxf(maxy, t[2 * j + 1]);
  }

  float tx1 = 0.25f * ((t[4] + t[6] + t[12] + t[14]) - (t[0] + t[2] + t[8] + t[10]));
  float ty1 = 0.25f * ((t[5] + t[7] + t[13] + t[15]) - (t[1] + t[3] + t[9] + t[11]));
  float tx2 = 0.25f * ((t[2] + t[6] + t[10] + t[14]) - (t[0] + t[4] + t[8] + t[12]));
  float ty2 = 0.25f * ((t[3] + t[7] + t[11] + t[15]) - (t[1] + t[5] + t[9] + t[13]));
  float tx3 = 0.25f * ((t[0] + t[2] + t[4] + t[6]) - (t[8] + t[10] + t[12] + t[14]));
  float ty3 = 0.25f * ((t[1] + t[3] + t[5] + t[7]) - (t[9] + t[11] + t[13] + t[15]));

  float* r = aux + (size_t)i * REC;
#pragma unroll
  for (int j = 0; j < 16; ++j) r[j] = t[j];
  r[16] = tx1; r[17] = ty1; r[18] = sqrtf(tx1 * tx1 + ty1 * ty1);
  r[19] = tx2; r[20] = ty2; r[21] = sqrtf(tx2 * tx2 + ty2 * ty2);
  r[22] = tx3; r[23] = ty3; r[24] = sqrtf(tx3 * tx3 + ty3 * ty3);
  r[25] = label;
  r[26] = valid ? 1.0f : 0.0f;
  r[27] = 0.0f;
  if (valid) {
    r[28] = minx; r[29] = miny; r[30] = maxx; r[31] = maxy;
  } else {
    // Degenerate (0,0,0,0) box: inter==0, areaB==0 => iou==0.  Output-
    // equivalent to the reference's -1 sentinel: an invalid box can never be
    // the argmax of a positive anchor, and neg-vs-ignore status is unchanged.
    r[28] = 0.0f; r[29] = 0.0f; r[30] = 0.0f; r[31] = 0.0f;
  }
}

__device__ __forceinline__ float smooth_l1(float d) {
  return (d <= (1.0f / 9.0f)) ? 4.5f * d * d : d - (1.0f / 18.0f);
}

#if FL_HAVE_TDM
// 1-D TDM descriptor: copy `tile_elems` f32 from gsrc into LDS at lds_off.
// tensor_elems clamps the valid extent: OOB reads land as zeros in LDS.
__device__ __forceinline__ void tdm_load_1d(const void* gsrc, unsigned lds_off,
                                            unsigned tile_elems,
                                            unsigned tensor_elems) {
  unsigned long long gaddr = (unsigned long long)gsrc;
  uint32x4 g0;
  g0[0] = 1u;                                       // count=1 (valid D#)
  g0[1] = lds_off;                                  // lds_addr (bytes)
  g0[2] = (unsigned)(gaddr & 0xFFFFFFFFull);        // global_addr[31:0]
  g0[3] = (unsigned)((gaddr >> 32) & 0x1FFFFFFull)  // global_addr[56:32]
          | (2u << 30);                             // type=2 ("image")
  int32x8 g1;
  g1[0] = (int)(2u << 16);                          // data_size=2 -> 4 bytes
  g1[1] = (int)((tensor_elems & 0xFFFFu) << 16);    // tensor_dim0[15:0]
  g1[2] = (int)((tensor_elems >> 16) | (1u << 16)); // dim0[31:16] | dim1=1
  g1[3] = (int)(tile_elems << 16);                  // dim1 hi=0 | tile_dim0
  g1[4] = (int)1u;                                  // tile_dim1=1, tile_dim2=0
  g1[5] = (int)tensor_elems;                        // tensor_dim0_stride lo32
  g1[6] = 0;
  g1[7] = 0;
  int32x4 z4 = {0, 0, 0, 0};
#if __clang_major__ >= 23
  int32x8 z8 = {0, 0, 0, 0, 0, 0, 0, 0};
  __builtin_amdgcn_tensor_load_to_lds(g0, g1, z4, z4, z8, 0);
#else
  __builtin_amdgcn_tensor_load_to_lds(g0, g1, z4, z4, 0);
#endif
}
#endif

// ---------------------------------------------------------------------------
// Phase 1: grid (NCHUNK, B); each block handles 1024 anchors of one image.
// ---------------------------------------------------------------------------
__global__ void __launch_bounds__(TPB)
main_kernel(const float* __restrict__ cls,
            const float* __restrict__ reg,
            const float* __restrict__ anchors,
            const float* __restrict__ aux,
            float* __restrict__ partials) {
  __shared__ __align__(16) float s_aux[M_ * REC];    // 16 KB annotation table
  __shared__ __align__(16) float s_anch[CHUNK * 4];  // 16 KB anchor slab
  __shared__ float4 s_red[TPB];                      // block reduction

  const int b     = blockIdx.y;
  const int chunk = blockIdx.x;
  const int tid   = threadIdx.x;
  const float* gsrc = aux + (size_t)b * M_ * REC;

  // Prefetch this thread's classification rows now; they are consumed only
  // after ~1800 VALU ops of IoU work, so the HBM latency is fully hidden.
#pragma unroll
  for (int k = 0; k < APT; ++k) {
    int a = chunk * CHUNK + k * TPB + tid;
    if (a < A_) __builtin_prefetch(cls + ((size_t)b * A_ + a) * C_, 0, 3);
  }

#if FL_HAVE_TDM
  if (tid < 32) {  // wave 0 issues two TDM DMAs; one wait covers both
    tdm_load_1d(gsrc, (unsigned)(unsigned long long)(void*)s_aux,
                (unsigned)(M_ * REC), (unsigned)(M_ * REC));
    tdm_load_1d(anchors + (size_t)chunk * CHUNK * 4,
                (unsigned)(unsigned long long)(void*)s_anch,
                (unsigned)(CHUNK * 4),
                (unsigned)(A_ * 4 - chunk * CHUNK * 4));  // tail -> zeros
    __builtin_amdgcn_s_wait_tensorcnt(0);
  }
  __syncthreads();
#else
  for (int i = tid; i < M_ * REC; i += TPB) s_aux[i] = gsrc[i];
  for (int i = tid; i < CHUNK * 4; i += TPB) {
    int gi = chunk * CHUNK * 4 + i;
    s_anch[i] = (gi < A_ * 4) ? anchors[gi] : 0.0f;
  }
  __syncthreads();
#endif

  const float S2[8] = {-1, -1, 1, 1, -1, -1, 1, 1};
  const float S4[8] = {-1, 1, -1, 1, -1, 1, -1, 1};
  const float S6[8] = {1, 1, 1, 1, -1, -1, -1, -1};

  float clsS = 0.0f, regS = 0.0f, vpS = 0.0f, cntS = 0.0f;

#pragma unroll
  for (int k = 0; k < APT; ++k) {
    int a = chunk * CHUNK + k * TPB + tid;
    bool active = (a < A_);

    float4 an = *reinterpret_cast<const float4*>(&s_anch[(k * TPB + tid) * 4]);
    float ax1 = an.x, ay1 = an.y, ax2 = an.z, ay2 = an.w;
    float aw = ax2 - ax1, ah = ay2 - ay1;
    float acx = ax1 + 0.5f * aw, acy = ay1 + 0.5f * ah;
    float areaA = aw * ah;

    // IoU argmax over 128 boxes: divide-free running max (strict '>' keeps
    // jnp.argmax first-occurrence semantics); both denominators > 0.
    float bn = -1.0f, bd = 1.0f;
    int bm = 0;
    for (int m = 0; m < M_; ++m) {
      float4 bb = *reinterpret_cast<const float4*>(&s_aux[m * REC + 28]);
      float iw = fminf(ax2, bb.z) - fmaxf(ax1, bb.x);
      float ih = fminf(ay2, bb.w) - fmaxf(ay1, bb.y);
      iw = fmaxf(iw, 0.0f);
      ih = fmaxf(ih, 0.0f);
      float inter = iw * ih;
      float areaB = (bb.z - bb.x) * (bb.w - bb.y);
      float den = fmaxf(areaA + areaB - inter, 1e-8f);
      bool c = inter * bd > bn * den;
      bn = c ? inter : bn;
      bd = c ? den : bd;
      bm = c ? m : bm;
    }
    // Threshold tests without the division: bd > 0 always.
    bool pos = (bn >= 0.5f * bd);
    bool neg = (bn < 0.4f * bd);

    const float* rec = &s_aux[bm * REC];
    int lab = pos ? (int)rec[25] : -1;

    if (active && (pos || neg)) {
      const float4* cp =
          reinterpret_cast<const float4*>(cls + ((size_t)b * A_ + a) * C_);
      float4 c0 = cp[0], c1 = cp[1];
      float pv[8] = {c0.x, c0.y, c0.z, c0.w, c1.x, c1.y, c1.z, c1.w};
#pragma unroll
      for (int j = 0; j < 8; ++j) {
        float p = fminf(fmaxf(pv[j], 1e-4f), 1.0f - 1e-4f);
        bool one = (j == lab);  // lab == -1 for negatives
        float w = one ? 0.25f * (1.0f - p) * (1.0f - p) : 0.75f * p * p;
        float l = one ? -logf(p) : -logf(1.0f - p);
        clsS += w * l;
      }
    }

    if (active && pos) {
      cntS += 1.0f;
      const float4* rp =
          reinterpret_cast<const float4*>(reg + ((size_t)b * A_ + a) * 8);
      float4 r0 = rp[0], r1 = rp[1];
      float r[8] = {r0.x, r0.y, r0.z, r0.w, r1.x, r1.y, r1.z, r1.w};
      float invW = 1.0f / aw, invH = 1.0f / ah;

      float rs = 0.0f;
#pragma unroll
      for (int j = 0; j < 8; ++j) {
        float px = r[0] + r[2] * S2[j] + r[4] * S4[j] + r[6] * S6[j];
        float py = r[1] + r[3] * S2[j] + r[5] * S4[j] + r[7] * S6[j];
        float tx = (rec[2 * j] - acx) * invW;
        float ty = (rec[2 * j + 1] - acy) * invH;
        rs += smooth_l1(fabsf(px - tx));
        rs += smooth_l1(fabsf(py - ty));
      }
      regS += rs * (1.0f / 16.0f);

      float vp = 0.0f;
#pragma unroll
      for (int i = 0; i < 3; ++i) {
        float rx = r[2 + 2 * i], ry = r[3 + 2 * i];
        float rn = sqrtf(rx * rx + ry * ry);
        float tx = rec[16 + 3 * i], ty = rec[17 + 3 * i], tn = rec[18 + 3 * i];
        vp += 1.0f - (rx * tx + ry * ty) / fmaxf(rn * tn, 1e-8f);
      }
      vpS += vp * (1.0f / 3.0f);
    }
  }

  // Deterministic block tree-reduction of (cls, reg, vp, pos_count).
  s_red[tid] = make_float4(clsS, regS, vpS, cntS);
  __syncthreads();
  for (int s = TPB / 2; s > 0; s >>= 1) {
    if (tid < s) {
      float4 x = s_red[tid], y = s_red[tid + s];
      s_red[tid] = make_float4(x.x + y.x, x.y + y.y, x.z + y.z, x.w + y.w);
    }
    __syncthreads();
  }
  if (tid == 0) {
    float4 v = s_red[0];
    float* p = partials + ((size_t)b * NCHUNK + chunk) * 4;
    p[0] = v.x; p[1] = v.y; p[2] = v.z; p[3] = v.w;
  }
}

// ---------------------------------------------------------------------------
// Phase 2: fixed-order reduction over chunks -> per-image losses -> mean.
// ---------------------------------------------------------------------------
__global__ void finalize_kernel(const float* __restrict__ partials,
                                float* __restrict__ out) {
  if (blockIdx.x == 0 && threadIdx.x == 0) {
    float mc = 0.0f, mr = 0.0f, mv = 0.0f;
    for (int b = 0; b < B_; ++b) {
      float sc = 0.0f, sr = 0.0f, sv = 0.0f, sp = 0.0f;
      for (int c = 0; c < NCHUNK; ++c) {
        const float* p = partials + ((size_t)b * NCHUNK + c) * 4;
        sc += p[0]; sr += p[1]; sv += p[2]; sp += p[3];
      }
      float np = fmaxf(sp, 1.0f);
      mc += sc / np;
      mr += sr / np;
      mv += sv / np;
    }
    out[0] = mc * (1.0f / B_);
    out[1] = mr * (1.0f / B_);
    out[2] = mv * (1.0f / B_);
  }
}

}  // namespace fl

extern "C" void kernel_launch(void* const* d_in, const int* in_sizes, int n_in,
                              void* d_out, int out_size, void* d_ws,
                              size_t ws_size, hipStream_t stream) {
  using namespace fl;
  (void)in_sizes; (void)n_in; (void)out_size; (void)ws_size;

  const float* cls     = (const float*)d_in[0];  // (B, A, C)
  const float* reg     = (const float*)d_in[1];  // (B, A, 8)
  const float* anchors = (const float*)d_in[2];  // (1, A, 4)
  const float* ann     = (const float*)d_in[3];  // (B, M, 18)
  // d_in[4] (embeddings) intentionally unused: it never enters the loss.

  float* aux      = (float*)d_ws;                // B*M*REC floats (128 KB)
  float* partials = aux + (size_t)B_ * M_ * REC; // B*NCHUNK*4 floats (~12 KB)

  phase0_kernel<<<(B_ * M_ + 255) / 256, 256, 0, stream>>>(ann, aux);
  main_kernel<<<dim3(NCHUNK, B_), TPB, 0, stream>>>(cls, reg, anchors, aux,
                                                    partials);
  finalize_kernel<<<1, 32, 0, stream>>>(partials, (float*)d_out);
}